// InvariantPointAttention_81862076661841
// MI455X (gfx1250) — compile-verified
//
#include <hip/hip_runtime.h>
#include <math.h>

typedef float v2f __attribute__((ext_vector_type(2)));
typedef float v8f __attribute__((ext_vector_type(8)));

#define NN 512
#define CS 384
#define CZ 128
#define CH 16
#define NH 12
#define PQ 4
#define PV 8
#define FIN 2112
#define W_L_C 0.5773502691896258f
#define W_C_HALF 0.11785113019775792f

// workspace layout (float offsets)
#define OFF_Q   0
#define OFF_K   (NN*192)
#define OFF_V   (2*NN*192)
#define OFF_QP  (3*NN*192)
#define OFF_KP  (OFF_QP + NN*144)
#define OFF_VP  (OFF_KP + NN*144)
#define OFF_FEAT (OFF_VP + NN*288)

__device__ __forceinline__ v8f wmma4(v2f a, v2f b, v8f c) {
  return __builtin_amdgcn_wmma_f32_16x16x4_f32(false, a, false, b,
                                               (short)0, c, false, false);
}

// ---------------------------------------------------------------------------
// Kernel 1: per-residue projections + rigid transform of point projections.
// ---------------------------------------------------------------------------
__global__ void ipa_proj(const float* __restrict__ s,
                         const float* __restrict__ rot,
                         const float* __restrict__ trans,
                         const float* __restrict__ Wq,
                         const float* __restrict__ Wk,
                         const float* __restrict__ Wv,
                         const float* __restrict__ Wqp,
                         const float* __restrict__ Wkp,
                         const float* __restrict__ Wvp,
                         float* __restrict__ ws) {
  __shared__ float srow[CS];
  int i = blockIdx.x;
  int tid = threadIdx.x;
  for (int c = tid; c < CS; c += blockDim.x) srow[c] = s[i*CS + c];
  __syncthreads();
  float* q  = ws + OFF_Q;  float* kk = ws + OFF_K;  float* v  = ws + OFF_V;
  float* qp = ws + OFF_QP; float* kp = ws + OFF_KP; float* vp = ws + OFF_VP;

  // q, k, v : (H*CH = 192 each)
  for (int o = tid; o < 3*NH*CH; o += blockDim.x) {
    int t = o / 192, rem = o % 192;
    const float* W = (t == 0) ? Wq : ((t == 1) ? Wk : Wv);
    float acc = 0.f;
    for (int c = 0; c < CS; ++c) acc += srow[c] * W[c*192 + rem];
    float* dst = (t == 0) ? q : ((t == 1) ? kk : v);
    dst[i*192 + rem] = acc;
  }
  // point projections: 48 qp + 48 kp + 96 vp points; each = 3 dots + rigid apply
  const float* R  = rot + i*9;
  const float* t3 = trans + i*3;
  for (int pt = tid; pt < 192; pt += blockDim.x) {
    const float* W; int off; float* dst; int stride;
    if (pt < 48)      { W = Wqp; off = pt*3;        dst = qp; stride = 144; }
    else if (pt < 96) { W = Wkp; off = (pt-48)*3;   dst = kp; stride = 144; }
    else              { W = Wvp; off = (pt-96)*3;   dst = vp; stride = 288; }
    float l0 = 0.f, l1 = 0.f, l2 = 0.f;
    for (int c = 0; c < CS; ++c) {
      float sv = srow[c];
      const float* wr = W + c*stride + off;
      l0 += sv*wr[0]; l1 += sv*wr[1]; l2 += sv*wr[2];
    }
    dst[i*stride + off + 0] = R[0]*l0 + R[1]*l1 + R[2]*l2 + t3[0];
    dst[i*stride + off + 1] = R[3]*l0 + R[4]*l1 + R[5]*l2 + t3[1];
    dst[i*stride + off + 2] = R[6]*l0 + R[7]*l1 + R[8]*l2 + t3[2];
  }
}

// ---------------------------------------------------------------------------
// Kernel 2: attention for one query residue i per block (4 waves, 128 thr).
// Dual-accumulator WMMA for pair bias and o_pair; VALU for per-head bits.
// ---------------------------------------------------------------------------
__global__ void ipa_attn(const float* __restrict__ z,
                         const float* __restrict__ rot,
                         const float* __restrict__ trans,
                         const float* __restrict__ Wb,
                         const float* __restrict__ gamma,
                         float* __restrict__ ws) {
  __shared__ float sm_a[NN*NH];     // logits -> attention weights (24 KB)
  __shared__ float sm_wb[CZ*16];    // Wb padded to 16 heads (8 KB)
  __shared__ float sm_q[NH*CH];
  __shared__ float sm_qp[NH*PQ*3];
  __shared__ float sm_g[NH];
  __shared__ float sm_s[NH];

  int i = blockIdx.x;
  int tid = threadIdx.x;
  int wave = tid >> 5, lane = tid & 31;
  int nl = lane & 15;                 // column-in-tile / M index
  int ks = (lane >> 4) * 2;           // K sub-offset for A/B fragments

  const float* q    = ws + OFF_Q;
  const float* kbuf = ws + OFF_K;
  const float* vbuf = ws + OFF_V;
  const float* qp   = ws + OFF_QP;
  const float* kp   = ws + OFF_KP;
  const float* vp   = ws + OFF_VP;
  float* feat = ws + OFF_FEAT + (size_t)i*FIN;
  const float* zi = z + (size_t)i*NN*CZ;

  for (int o = tid; o < CZ*16; o += blockDim.x) {
    int c = o >> 4, h = o & 15;
    sm_wb[o] = (h < NH) ? Wb[c*NH + h] : 0.f;
  }
  for (int o = tid; o < NH*CH;   o += blockDim.x) sm_q[o]  = q[i*192 + o];
  for (int o = tid; o < NH*PQ*3; o += blockDim.x) sm_qp[o] = qp[i*144 + o];
  if (tid < NH) sm_g[tid] = gamma[tid];
  __syncthreads();

  // ---- pair bias b[j,h] = sum_c z[i,j,c] * Wb[c,h] via WMMA ----
  for (int tile = wave; tile < NN/16; tile += 4) {
    int j0 = tile*16;
    v8f acc0 = {}, acc1 = {};
    const float* za = zi + (size_t)(j0 + nl)*CZ + ks;   // A[m=j][k]
    for (int k4 = 0; k4 < CZ; k4 += 8) {
      v2f a0; a0.x = za[k4];     a0.y = za[k4+1];
      v2f b0; b0.x = sm_wb[(k4+ks)*16 + nl];   b0.y = sm_wb[(k4+ks+1)*16 + nl];
      v2f a1; a1.x = za[k4+4];   a1.y = za[k4+5];
      v2f b1; b1.x = sm_wb[(k4+4+ks)*16 + nl]; b1.y = sm_wb[(k4+5+ks)*16 + nl];
      acc0 = wmma4(a0, b0, acc0);
      acc1 = wmma4(a1, b1, acc1);
    }
    v8f acc = acc0 + acc1;
    int h = nl;
    if (h < NH) {
      int jb = j0 + ((lane >> 4) ? 8 : 0);
      for (int r = 0; r < 8; ++r) sm_a[(jb + r)*NH + h] = acc[r];
    }
  }
  __syncthreads();

  // ---- logits = W_L * (qk/sqrt(16) + b - (W_C/2)*gamma*d2) ----
  for (int idx = tid; idx < NN*NH; idx += blockDim.x) {
    int j = idx / NH, h = idx % NH;
    const float* kj = kbuf + j*192 + h*16;
    const float* qh = sm_q + h*16;
    float qk = 0.f;
    for (int d = 0; d < 16; ++d) qk += qh[d]*kj[d];
    qk *= 0.25f;
    const float* kpj = kp + j*144 + h*12;
    const float* qph = sm_qp + h*12;
    float d2 = 0.f;
    for (int e = 0; e < 12; ++e) { float df = qph[e]-kpj[e]; d2 += df*df; }
    sm_a[idx] = W_L_C * (qk + sm_a[idx] - W_C_HALF * sm_g[h] * d2);
  }
  __syncthreads();

  // ---- softmax over j (12 heads handled by 12 threads) ----
  if (tid < NH) {
    int h = tid;
    float m = -1e30f;
    for (int j = 0; j < NN; ++j) m = fmaxf(m, sm_a[j*NH + h]);
    float ssum = 0.f;
    for (int j = 0; j < NN; ++j) {
      float e = __expf(sm_a[j*NH + h] - m);
      sm_a[j*NH + h] = e;
      ssum += e;
    }
    sm_s[h] = 1.f / ssum;
  }
  __syncthreads();
  for (int idx = tid; idx < NN*NH; idx += blockDim.x) sm_a[idx] *= sm_s[idx % NH];
  __syncthreads();

  // ---- o_pair[h,c] = sum_j a[j,h] * z[i,j,c] via WMMA (M=h, N=c, K=j) ----
  bool hok = (nl < NH);
  for (int ct = wave; ct < CZ/16; ct += 4) {
    int c0 = ct*16;
    v8f acc0 = {}, acc1 = {};
    for (int j0 = 0; j0 < NN; j0 += 8) {
      const float* zb = zi + (size_t)(j0 + ks)*CZ + c0 + nl;
      // prefetch the z rows 16 j-steps ahead (streaming operand)
      __builtin_prefetch(zb + 16*CZ, 0, 0);
      v2f a0, a1;
      a0.x = hok ? sm_a[(j0+ks)*NH + nl]   : 0.f;
      a0.y = hok ? sm_a[(j0+ks+1)*NH + nl] : 0.f;
      a1.x = hok ? sm_a[(j0+4+ks)*NH + nl]   : 0.f;
      a1.y = hok ? sm_a[(j0+5+ks)*NH + nl] : 0.f;
      v2f b0; b0.x = zb[0];      b0.y = zb[CZ];
      v2f b1; b1.x = zb[4*CZ];   b1.y = zb[5*CZ];
      acc0 = wmma4(a0, b0, acc0);
      acc1 = wmma4(a1, b1, acc1);
    }
    v8f acc = acc0 + acc1;
    int hbase = (lane >> 4) ? 8 : 0;
    for (int r = 0; r < 8; ++r) {
      int h = hbase + r;
      if (h < NH) feat[h*CZ + c0 + nl] = acc[r];
    }
  }

  // ---- o_sc[h,d] = sum_j a[j,h] * v[j,h,d] ----
  for (int o = tid; o < NH*CH; o += blockDim.x) {
    int h = o >> 4;
    float acc = 0.f;
    for (int j = 0; j < NN; ++j) acc += sm_a[j*NH + h] * vbuf[j*192 + o];
    feat[1536 + o] = acc;
  }

  // ---- o_pt: accumulate global, inverse rigid, norm ----
  const float* R  = rot + i*9;
  const float* t3 = trans + i*3;
  for (int pt = tid; pt < NH*PV; pt += blockDim.x) {
    int h = pt >> 3, p = pt & 7;
    int off = h*24 + p*3;
    float g0 = 0.f, g1 = 0.f, g2 = 0.f;
    for (int j = 0; j < NN; ++j) {
      float a = sm_a[j*NH + h];
      const float* vg = vp + j*288 + off;
      g0 += a*vg[0]; g1 += a*vg[1]; g2 += a*vg[2];
    }
    g0 -= t3[0]; g1 -= t3[1]; g2 -= t3[2];
    float l0 = R[0]*g0 + R[3]*g1 + R[6]*g2;   // R^T (x - t)
    float l1 = R[1]*g0 + R[4]*g1 + R[7]*g2;
    float l2 = R[2]*g0 + R[5]*g1 + R[8]*g2;
    feat[1728 + off + 0] = l0;
    feat[1728 + off + 1] = l1;
    feat[1728 + off + 2] = l2;
    feat[2016 + h*8 + p] = sqrtf(l0*l0 + l1*l1 + l2*l2 + 1e-12f);
  }
}

// ---------------------------------------------------------------------------
// Kernel 3: out = feat(512x2112) @ Wf(2112x384) + bf, tiled WMMA f32 16x16x4.
// One 16x16 output tile per wave; 768 tiles -> 192 blocks x 4 waves.
// ---------------------------------------------------------------------------
__global__ void ipa_out(const float* __restrict__ feat,
                        const float* __restrict__ Wf,
                        const float* __restrict__ bf,
                        float* __restrict__ out) {
  int tid = threadIdx.x;
  int wave = tid >> 5, lane = tid & 31;
  int tile = blockIdx.x*4 + wave;        // 0..767
  int row0 = (tile / 24) * 16;
  int col0 = (tile % 24) * 16;
  int nl = lane & 15;
  int ks = (lane >> 4) * 2;

  v8f acc0 = {}, acc1 = {};
  const float* arow = feat + (size_t)(row0 + nl)*FIN + ks;
  const float* bcol = Wf + (size_t)ks*384 + col0 + nl;
  for (int k4 = 0; k4 < FIN; k4 += 8) {
    // prefetch Wf rows 32 k-steps ahead (strided stream)
    __builtin_prefetch(bcol + (size_t)(k4 + 32)*384, 0, 0);
    v2f a0; a0.x = arow[k4];     a0.y = arow[k4+1];
    v2f b0; b0.x = bcol[(size_t)k4*384];     b0.y = bcol[(size_t)(k4+1)*384];
    v2f a1; a1.x = arow[k4+4];   a1.y = arow[k4+5];
    v2f b1; b1.x = bcol[(size_t)(k4+4)*384]; b1.y = bcol[(size_t)(k4+5)*384];
    acc0 = wmma4(a0, b0, acc0);
    acc1 = wmma4(a1, b1, acc1);
  }
  v8f acc = acc0 + acc1;
  int rbase = (lane >> 4) ? 8 : 0;
  float bias = bf[col0 + nl];
  for (int r = 0; r < 8; ++r)
    out[(size_t)(row0 + rbase + r)*384 + col0 + nl] = acc[r] + bias;
}

// ---------------------------------------------------------------------------
extern "C" void kernel_launch(void* const* d_in, const int* in_sizes, int n_in,
                              void* d_out, int out_size, void* d_ws, size_t ws_size,
                              hipStream_t stream) {
  (void)in_sizes; (void)n_in; (void)out_size; (void)ws_size;
  const float* s     = (const float*)d_in[0];
  const float* z     = (const float*)d_in[1];
  const float* rot   = (const float*)d_in[2];
  const float* trans = (const float*)d_in[3];
  const float* Wq    = (const float*)d_in[4];
  const float* Wk    = (const float*)d_in[5];
  const float* Wv    = (const float*)d_in[6];
  const float* Wqp   = (const float*)d_in[7];
  const float* Wkp   = (const float*)d_in[8];
  const float* Wvp   = (const float*)d_in[9];
  const float* Wb    = (const float*)d_in[10];
  const float* gamma = (const float*)d_in[11];
  const float* Wf    = (const float*)d_in[12];
  const float* bf    = (const float*)d_in[13];
  float* out = (float*)d_out;
  float* ws  = (float*)d_ws;

  ipa_proj<<<NN, 256, 0, stream>>>(s, rot, trans, Wq, Wk, Wv, Wqp, Wkp, Wvp, ws);
  ipa_attn<<<NN, 128, 0, stream>>>(z, rot, trans, Wb, gamma, ws);
  ipa_out<<<192, 128, 0, stream>>>(ws + OFF_FEAT, Wf, bf, out);
}